// GPT2PagedAttention_18365280157696
// MI455X (gfx1250) — compile-verified
//
#include <hip/hip_runtime.h>
#include <hip/hip_bf16.h>
#include <math.h>
#include <stdint.h>

typedef __attribute__((ext_vector_type(2))) float v2f;
typedef __attribute__((ext_vector_type(8))) float v8f;
typedef __attribute__((ext_vector_type(4))) unsigned int v4u;
typedef __attribute__((ext_vector_type(4))) int v4i;
typedef __attribute__((ext_vector_type(8))) int v8i;

#define BB   8
#define QQ   16
#define DD   1024
#define HH   16
#define DHH  64
#define PP   32
#define PSS  128
#define SPLIT 8
#define PAGES_PER_SPLIT (PP / SPLIT)
#define PART_STRIDE 1056   // 16*64 acc + 16 m + 16 l
#define CHUNK_ELEMS (16 * DHH)   // one 16-key x 64-dh fp32 chunk = 4KB

__device__ __forceinline__ v8f wmma_f32_k4(v2f a, v2f b, v8f c) {
  // D = A(16x4 f32) * B(4x16 f32) + C(16x16 f32)
  return __builtin_amdgcn_wmma_f32_16x16x4_f32(false, a, false, b, (short)0, c,
                                               false, false);
}

// ---------------------------------------------------------------------------
// TDM: DMA one contiguous 4KB fp32 chunk (1024 elems) global -> LDS.
// D# group0: count=1, lds_addr, 57b global addr, type=2.
// D# group1: data_size=4B, tensor_dim0=tile_dim0=1024, dim1=1, stride0=1024.
// Tracked by TENSORcnt; EXEC ignored (per-wave DMA).
// ---------------------------------------------------------------------------
__device__ __forceinline__ void tdm_load_chunk(const float* gsrc,
                                               uint32_t lds_byte_addr) {
  const uint64_t ga = (uint64_t)(uintptr_t)gsrc;
  v4u g0 = { 1u,                                             // count=1
             lds_byte_addr,                                  // lds_addr
             (uint32_t)ga,                                   // global_addr lo
             (uint32_t)((ga >> 32) & 0x1FFFFFFu) | (2u << 30) }; // hi + type=2
  v8i g1 = { 0x20000,      // data_size=2 (4 bytes); workgroup_mask=0
             0x04000000,   // tensor_dim0 = 1024 (low16 at bits[31:16])
             0x00010000,   // tensor_dim1 = 1
             0x04000000,   // tile_dim0   = 1024
             0x00000001,   // tile_dim1   = 1
             1024,         // tensor_dim0_stride lo
             0, 0 };
  v4i z4 = { 0, 0, 0, 0 };
#if __clang_major__ >= 23
  v8i z8 = { 0, 0, 0, 0, 0, 0, 0, 0 };
  __builtin_amdgcn_tensor_load_to_lds(g0, g1, z4, z4, z8, 0);
#else
  __builtin_amdgcn_tensor_load_to_lds(g0, g1, z4, z4, 0);
#endif
}

// ---------------------------------------------------------------------------
// Kernel 1: QKV projection. One wave per 16x16 output tile.
// ---------------------------------------------------------------------------
__global__ __launch_bounds__(32) void qkv_kernel(
    const float* __restrict__ hidden, const float* __restrict__ w,
    const float* __restrict__ bias,
    float* __restrict__ qo, float* __restrict__ ko, float* __restrict__ vo)
{
  const int tm   = blockIdx.x;           // 0..7
  const int tn   = blockIdx.y;           // 0..191
  const int lane = threadIdx.x & 31;
  const int half = lane >> 4, ln = lane & 15;
  const int col  = tn * 16 + ln;

  const float* ar = hidden + ((size_t)(tm * 16 + ln)) * DD + half * 2;
  v8f c = {};
  for (int k0 = 0; k0 < DD; k0 += 4) {
    v2f a = *(const v2f*)(ar + k0);
    const int kb = k0 + half * 2;
    v2f bf = { w[(size_t)kb * (3 * DD) + col],
               w[(size_t)(kb + 1) * (3 * DD) + col] };
    c = wmma_f32_k4(a, bf, c);
  }
  const float bv   = bias[col];
  const int which  = tn / 64;            // 0=q 1=k 2=v
  const int cc     = col - which * DD;
  const int hh     = cc / DHH, dh = cc % DHH;
  float* dst = (which == 0) ? qo : (which == 1) ? ko : vo;
#pragma unroll
  for (int j = 0; j < 8; ++j) {
    const int r  = tm * 16 + j + half * 8;
    const int b  = r >> 4, qi = r & 15;
    dst[(((size_t)b * HH + hh) * QQ + qi) * DHH + dh] = c[j] + bv;
  }
}

// ---------------------------------------------------------------------------
// Flash-attention state + one 16-key chunk (WMMA QK^T, online softmax,
// LDS transpose of probs, WMMA PV). kchunk/vchunk point at a [16][64] tile
// (LDS in the partial kernel, global in the combine kernel); kofs is the
// chunk's key offset within the page for masking.
// ---------------------------------------------------------------------------
struct AttnState { float m[8]; float l[8]; v8f acc[4]; };

__device__ __forceinline__ void attn_chunk(
    const v2f (&aq)[16], const float* kchunk, const float* vchunk,
    int kofs, int start, int len, bool causal,
    int half, int ln, float* lds_e, AttnState& st)
{
  // --- S = q(16x64) x k^T(64x16), 16 WMMA steps over DH ---
  v8f s = {};
  const float* kb = kchunk + (size_t)ln * DHH + half * 2;
#pragma unroll
  for (int kk = 0; kk < 16; ++kk) {
    v2f bf = *(const v2f*)(kb + 4 * kk);
    s = wmma_f32_k4(aq[kk], bf, s);
  }

  const int   kidx  = kofs + ln;                       // key column index
  const float cmask = ((kidx >= start) && (kidx < start + len)) ? 0.f
                                                                : -INFINITY;
  float e8[8];
#pragma unroll
  for (int j = 0; j < 8; ++j) {
    const int row = j + half * 8;
    float add = cmask;
    if (causal && kidx > row) add = -INFINITY;
    float sv = s[j] * 0.125f + add;                    // scale = 1/sqrt(64)
    float rm = sv;
    rm = fmaxf(rm, __shfl_xor(rm, 1, 32));
    rm = fmaxf(rm, __shfl_xor(rm, 2, 32));
    rm = fmaxf(rm, __shfl_xor(rm, 4, 32));
    rm = fmaxf(rm, __shfl_xor(rm, 8, 32));
    const float mn = fmaxf(st.m[j], rm);
    const float al = (mn == -INFINITY) ? 1.f : __expf(st.m[j] - mn);
    const float e  = (sv == -INFINITY) ? 0.f : __expf(sv - mn);
    float rs = e;
    rs += __shfl_xor(rs, 1, 32);
    rs += __shfl_xor(rs, 2, 32);
    rs += __shfl_xor(rs, 4, 32);
    rs += __shfl_xor(rs, 8, 32);
    st.l[j] = al * st.l[j] + rs;
    st.m[j] = mn;
    e8[j]   = e;
#pragma unroll
    for (int t = 0; t < 4; ++t) st.acc[t][j] = al * st.acc[t][j];
  }

  // --- transpose probs C-layout -> A-layout via LDS ---
#pragma unroll
  for (int j = 0; j < 8; ++j) lds_e[(j + half * 8) * 16 + ln] = e8[j];
  __syncthreads();

  // --- acc += P(16x16) x V(16x64), 4 DH tiles x 4 K-steps ---
#pragma unroll
  for (int kk = 0; kk < 4; ++kk) {
    v2f ea = *(const v2f*)(&lds_e[ln * 16 + 4 * kk + half * 2]);
    const float* vb0 = vchunk + (size_t)(4 * kk + half * 2) * DHH;
#pragma unroll
    for (int t = 0; t < 4; ++t) {
      v2f vb = { vb0[t * 16 + ln], vb0[DHH + t * 16 + ln] };
      st.acc[t] = wmma_f32_k4(ea, vb, st.acc[t]);
    }
  }
  __syncthreads();
}

// ---------------------------------------------------------------------------
// Kernel 2a: split-K paged pass with TDM double-buffered KV staging.
// grid = (B*H, SPLIT); each wave streams PAGES_PER_SPLIT pages.
// ---------------------------------------------------------------------------
__global__ __launch_bounds__(32) void attn_partial_kernel(
    const float* __restrict__ qh, const float* __restrict__ k_pages,
    const float* __restrict__ v_pages, const int* __restrict__ page_pos,
    float* __restrict__ part)
{
  const int bh = blockIdx.x, sp = blockIdx.y;
  const int b = bh / HH, h = bh % HH;
  const int lane = threadIdx.x & 31;
  const int half = lane >> 4, ln = lane & 15;
  __shared__ float ldsK[2][CHUNK_ELEMS];
  __shared__ float ldsV[2][CHUNK_ELEMS];
  __shared__ float lds_e[QQ * 16];

  const float* qrow = qh + ((size_t)(b * HH + h) * QQ + ln) * DHH + half * 2;
  v2f aq[16];
#pragma unroll
  for (int s = 0; s < 16; ++s) aq[s] = *(const v2f*)(qrow + 4 * s);

  AttnState st;
#pragma unroll
  for (int j = 0; j < 8; ++j) { st.m[j] = -INFINITY; st.l[j] = 0.f; }
  v8f z = {};
#pragma unroll
  for (int t = 0; t < 4; ++t) st.acc[t] = z;

  // chunk c (0..31): page sp*4 + c/8, key window (c%8)*16
  auto chunk_src = [&](int c, const float* pages) -> const float* {
    const int p = sp * PAGES_PER_SPLIT + (c >> 3);
    const size_t pbase = (((size_t)p * BB + b) * HH + h) * PSS * DHH;
    return pages + pbase + (size_t)(c & 7) * CHUNK_ELEMS;
  };

  const int NCH = PAGES_PER_SPLIT * (PSS / 16);   // 32 chunks per split
  tdm_load_chunk(chunk_src(0, k_pages), (uint32_t)(uintptr_t)&ldsK[0][0]);
  tdm_load_chunk(chunk_src(0, v_pages), (uint32_t)(uintptr_t)&ldsV[0][0]);

  int buf = 0;
  for (int c = 0; c < NCH; ++c) {
    if (c + 1 < NCH) {
      // issue next chunk's DMA into the other buffer, then wait for the two
      // oldest (this chunk's) TDM ops: TENSORcnt is in-order per wave.
      tdm_load_chunk(chunk_src(c + 1, k_pages),
                     (uint32_t)(uintptr_t)&ldsK[buf ^ 1][0]);
      tdm_load_chunk(chunk_src(c + 1, v_pages),
                     (uint32_t)(uintptr_t)&ldsV[buf ^ 1][0]);
      __builtin_amdgcn_s_wait_tensorcnt((short)2);
    } else {
      __builtin_amdgcn_s_wait_tensorcnt((short)0);
    }
    if ((c & 7) == 0 && c + 8 < NCH)             // nudge next page toward L2
      __builtin_prefetch(chunk_src(c + 8, k_pages), 0, 1);

    const int p     = sp * PAGES_PER_SPLIT + (c >> 3);
    const int start = page_pos[(p * BB + b) * 2 + 0];
    const int len   = page_pos[(p * BB + b) * 2 + 1];
    attn_chunk(aq, &ldsK[buf][0], &ldsV[buf][0], (c & 7) * 16,
               start, len, false, half, ln, lds_e, st);
    buf ^= 1;
  }

  float* pb = part + ((size_t)bh * SPLIT + sp) * PART_STRIDE;
#pragma unroll
  for (int j = 0; j < 8; ++j) {
    const int row = j + half * 8;
#pragma unroll
    for (int t = 0; t < 4; ++t)
      pb[row * DHH + t * 16 + ln] = st.acc[t][j];
    if (ln == 0) {
      pb[1024 + row] = st.m[j];
      pb[1040 + row] = st.l[j];
    }
  }
}

// ---------------------------------------------------------------------------
// Kernel 2b: merge SPLIT partials, run causal current-token tile (WMMA),
// normalize and emit ctx in [B,Q,D] layout.
// ---------------------------------------------------------------------------
__global__ __launch_bounds__(32) void attn_combine_kernel(
    const float* __restrict__ qh, const float* __restrict__ kcur,
    const float* __restrict__ vcur, const int* __restrict__ input_pos,
    const float* __restrict__ part, float* __restrict__ ctx)
{
  const int bh = blockIdx.x;
  const int b = bh / HH, h = bh % HH;
  const int lane = threadIdx.x & 31;
  const int half = lane >> 4, ln = lane & 15;
  __shared__ float lds_e[QQ * 16];

  AttnState st;
  v8f z = {};
#pragma unroll
  for (int t = 0; t < 4; ++t) st.acc[t] = z;
#pragma unroll
  for (int j = 0; j < 8; ++j) { st.m[j] = -INFINITY; st.l[j] = 0.f; }

  const float* base = part + (size_t)bh * SPLIT * PART_STRIDE;
  for (int s = 0; s < SPLIT; ++s) {
    const float* pb = base + (size_t)s * PART_STRIDE;
#pragma unroll
    for (int j = 0; j < 8; ++j)
      st.m[j] = fmaxf(st.m[j], pb[1024 + j + half * 8]);
  }
  for (int s = 0; s < SPLIT; ++s) {
    const float* pb = base + (size_t)s * PART_STRIDE;
#pragma unroll
    for (int j = 0; j < 8; ++j) {
      const int row = j + half * 8;
      const float pm = pb[1024 + row];
      const float wt = (pm == -INFINITY) ? 0.f : __expf(pm - st.m[j]);
      st.l[j] += wt * pb[1040 + row];
#pragma unroll
      for (int t = 0; t < 4; ++t)
        st.acc[t][j] += wt * pb[row * DHH + t * 16 + ln];
    }
  }

  // current-token causal tile (global K/V, single chunk)
  const float* qrow = qh + ((size_t)(b * HH + h) * QQ + ln) * DHH + half * 2;
  v2f aq[16];
#pragma unroll
  for (int s = 0; s < 16; ++s) aq[s] = *(const v2f*)(qrow + 4 * s);
  const float* kp = kcur + (size_t)(b * HH + h) * QQ * DHH;
  const float* vp = vcur + (size_t)(b * HH + h) * QQ * DHH;
  const int len = input_pos[b * 2 + 1];
  attn_chunk(aq, kp, vp, 0, 0, len, true, half, ln, lds_e, st);

  // normalize + write ctx [B,Q,D]
#pragma unroll
  for (int j = 0; j < 8; ++j) {
    const int row = j + half * 8;
    const float inv = 1.f / fmaxf(st.l[j], 1e-9f);
#pragma unroll
    for (int t = 0; t < 4; ++t)
      ctx[((size_t)(b * QQ + row)) * DD + h * DHH + t * 16 + ln] =
          st.acc[t][j] * inv;
  }
}

// ---------------------------------------------------------------------------
// Kernel 3: output projection  out = ctx @ c_proj_w + c_proj_b
// ---------------------------------------------------------------------------
__global__ __launch_bounds__(32) void proj_kernel(
    const float* __restrict__ ctx, const float* __restrict__ w,
    const float* __restrict__ bias, float* __restrict__ out)
{
  const int tm = blockIdx.x;             // 0..7
  const int tn = blockIdx.y;             // 0..63
  const int lane = threadIdx.x & 31;
  const int half = lane >> 4, ln = lane & 15;
  const int col = tn * 16 + ln;

  const float* ar = ctx + ((size_t)(tm * 16 + ln)) * DD + half * 2;
  v8f c = {};
  for (int k0 = 0; k0 < DD; k0 += 4) {
    v2f a = *(const v2f*)(ar + k0);
    const int kb = k0 + half * 2;
    v2f bf = { w[(size_t)kb * DD + col], w[(size_t)(kb + 1) * DD + col] };
    c = wmma_f32_k4(a, bf, c);
  }
  const float bv = bias[col];
#pragma unroll
  for (int j = 0; j < 8; ++j) {
    const int r = tm * 16 + j + half * 8;
    out[(size_t)r * DD + col] = c[j] + bv;
  }
}

extern "C" void kernel_launch(void* const* d_in, const int* in_sizes, int n_in,
                              void* d_out, int out_size, void* d_ws,
                              size_t ws_size, hipStream_t stream) {
  const float* hidden   = (const float*)d_in[0];
  const float* k_pages  = (const float*)d_in[1];
  const float* v_pages  = (const float*)d_in[2];
  const float* c_attn_w = (const float*)d_in[3];
  const float* c_attn_b = (const float*)d_in[4];
  const float* c_proj_w = (const float*)d_in[5];
  const float* c_proj_b = (const float*)d_in[6];
  const int*   page_pos = (const int*)d_in[7];
  const int*   input_pos= (const int*)d_in[8];
  float* out = (float*)d_out;

  float* ws = (float*)d_ws;
  const size_t HEADS = (size_t)BB * HH * QQ * DHH;   // 131072 floats
  float* q    = ws;
  float* k    = ws + HEADS;
  float* v    = ws + 2 * HEADS;
  float* ctx  = ws + 3 * HEADS;                      // B*Q*D floats
  float* part = ws + 4 * HEADS;                      // 128*8*1056 floats

  qkv_kernel<<<dim3(8, 192), 32, 0, stream>>>(hidden, c_attn_w, c_attn_b,
                                              q, k, v);
  attn_partial_kernel<<<dim3(BB * HH, SPLIT), 32, 0, stream>>>(
      q, k_pages, v_pages, page_pos, part);
  attn_combine_kernel<<<dim3(BB * HH), 32, 0, stream>>>(
      q, k, v, input_pos, part, ctx);
  proj_kernel<<<dim3(8, 64), 32, 0, stream>>>(ctx, c_proj_w, c_proj_b, out);
}